// MultiInnerProductDecoder_15264313770417
// MI455X (gfx1250) — compile-verified
//
#include <hip/hip_runtime.h>

typedef __attribute__((ext_vector_type(2))) float v2f;
typedef __attribute__((ext_vector_type(8))) float v8f;

#define IN_DIM        128
#define E_PER_ETYPE   131072
#define NUM_ET_USED   16
// 16 edges per tile -> 8192 tiles per edge type, 131072 tiles total.

__global__ __launch_bounds__(256) void mipd_wmma_kernel(
    const float* __restrict__ z,            // (100000, 128) f32
    const float* __restrict__ weight,       // (18, 128) f32 (only first 16 rows used)
    const long long* __restrict__ edge_index, // (16, 2, 131072) i64
    float* __restrict__ out)                // (16, 131072) f32
{
  const int lane        = threadIdx.x & 31;
  const int waveInBlock = threadIdx.x >> 5;
  const int tile        = blockIdx.x * 8 + waveInBlock; // one 16-edge tile per wave32

  const int t  = tile >> 13;           // tile / 8192  -> edge type
  const int e0 = (tile & 8191) << 4;   // *16          -> first edge of tile

  const int m = lane & 15;             // A-row / B-col: which edge in the tile
  const int h = lane >> 4;             // half-wave -> K sub-offset (ISA 16x4 / 4x16 layout)

  // Gather the two row indices for this lane's edge (both half-waves load the
  // same values; addresses coalesce within 128B lines).
  const long long src = edge_index[(long long)(t * 2 + 0) * E_PER_ETYPE + e0 + m];
  const long long dst = edge_index[(long long)(t * 2 + 1) * E_PER_ETYPE + e0 + m];

  const float* zs = z + src * IN_DIM;
  const float* zd = z + dst * IN_DIM;
  const float* wr = weight + t * IN_DIM;

  v8f c = {0.f, 0.f, 0.f, 0.f, 0.f, 0.f, 0.f, 0.f};

  // D = A x B over K=128 in 32 chained fp32 WMMAs (K=4 each).
  // A[m][k]   = zs_m[k]            (16x4 fragment: lane m/h holds k = 4s+2h, 4s+2h+1)
  // B[k][m]   = zd_m[k] * w[t][k]  (4x16 fragment: identical per-lane k offsets)
  // D[i][i]   = desired dot product for edge i.
#pragma unroll
  for (int s = 0; s < 32; ++s) {
    const int k = 4 * s + 2 * h;
    v2f a  = *(const v2f*)(zs + k);
    v2f b  = *(const v2f*)(zd + k);
    v2f w2 = *(const v2f*)(wr + k);
    b.x *= w2.x;
    b.y *= w2.y;
    c = __builtin_amdgcn_wmma_f32_16x16x4_f32(
        /*neg_a=*/false, a, /*neg_b=*/false, b,
        /*c_mod=*/(short)0, c, /*reuse_a=*/false, /*reuse_b=*/false);
  }

  // Diagonal extraction (16x16 f32 C/D layout, ISA 7.12.2):
  //   D[i][i] for i<8  -> lane i,    VGPR i
  //   D[i][i] for i>=8 -> lane i+16, VGPR i-8
  // Holder lanes: 0..7 and 24..31, accumulator index = lane & 7.
  const int idx = lane & 7;
  float v0 = (idx & 1) ? c[1] : c[0];
  float v1 = (idx & 1) ? c[3] : c[2];
  float v2 = (idx & 1) ? c[5] : c[4];
  float v3 = (idx & 1) ? c[7] : c[6];
  float u0 = (idx & 2) ? v1 : v0;
  float u1 = (idx & 2) ? v3 : v2;
  float val = (idx & 4) ? u1 : u0;

  const bool valid = (lane < 8) || (lane >= 24);
  const int  i     = (lane < 8) ? lane : (lane - 16);
  if (valid) {
    const float sig = 1.0f / (1.0f + __expf(-val));
    out[(long long)t * E_PER_ETYPE + e0 + i] = sig;
  }
}

extern "C" void kernel_launch(void* const* d_in, const int* in_sizes, int n_in,
                              void* d_out, int out_size, void* d_ws, size_t ws_size,
                              hipStream_t stream) {
  const float*     z          = (const float*)d_in[0];
  const float*     weight     = (const float*)d_in[1];
  const long long* edge_index = (const long long*)d_in[2];
  float*           out        = (float*)d_out;

  // 131072 tiles total, 8 waves (tiles) per 256-thread block -> 16384 blocks.
  dim3 grid(16384), block(256);
  hipLaunchKernelGGL(mipd_wmma_kernel, grid, block, 0, stream,
                     z, weight, edge_index, out);
}